// NCI1GCN_40321152974882
// MI455X (gfx1250) — compile-verified
//
#include <hip/hip_runtime.h>

#define N_NODES  100000
#define N_EDGES  1600000
#define N_GRAPHS 2000
#define IN_DIM   37
#define HID      128
#define OUT_CLS  2
#define DPAD0    40      // IN_DIM rounded up to multiple of 4

typedef __attribute__((ext_vector_type(2))) float v2f;
typedef __attribute__((ext_vector_type(8))) float v8f;

// ---------------------------------------------------------------- utilities
__global__ void zero_kernel(float* __restrict__ p, int n) {
    int i = blockIdx.x * blockDim.x + threadIdx.x;
    if (i < n) p[i] = 0.0f;
}

// deg[i] = number of incoming edges (edge weights are all 1.0) — layer-invariant
__global__ void deg_kernel(const long long* __restrict__ ei, float* __restrict__ deg) {
    int e = blockIdx.x * blockDim.x + threadIdx.x;
    if (e < N_EDGES) {
        int dst = (int)ei[N_EDGES + e];
        atomicAdd(&deg[dst], 1.0f);
    }
}

// -------------------------------------------- weight pre-pack into B-fragments
// P is indexed in float2 units: j = (k0/4)*256 + half*128 + col, holding
// {W[k0+2*half][col], W[k0+2*half+1][col]}, zero-padded for k >= DI.
// A wave then fetches its whole B fragment with one coalesced global_load_b64.
template <int DI, int DPAD>
__global__ void pack_w_kernel(const float* __restrict__ W, float* __restrict__ P) {
    int j = blockIdx.x * blockDim.x + threadIdx.x;   // float2 index
    if (j >= DPAD * 64) return;
    const int col = j & 127;
    const int t   = j >> 7;                          // (k0/4)*2 + half
    const int k   = (t >> 1) * 4 + (t & 1) * 2;      // = k0 + kb
    const float lo = (k < DI)     ? W[(size_t)k * HID + col]       : 0.0f;
    const float hi = (k + 1 < DI) ? W[(size_t)(k + 1) * HID + col] : 0.0f;
    reinterpret_cast<float2*>(P)[j] = make_float2(lo, hi);
}

// ------------------------------------------------ LEConv fused triple GEMM
// acc1 = x@W1, acc2 = x@W2, acc3 = x@W3 via V_WMMA_F32_16X16X4_F32.
// Epilogue: ac = deg .* (acc1+b1) + (acc3+b3);  bb = acc2.
// Block = 256 threads = 8 waves; wave w owns output columns [16w, 16w+16).
template <int DI, int DPAD>
__global__ __launch_bounds__(256)
void leconv_gemm_kernel(const float* __restrict__ hin,
                        const float* __restrict__ P1, const float* __restrict__ b1,
                        const float* __restrict__ P2,
                        const float* __restrict__ P3, const float* __restrict__ b3,
                        const float* __restrict__ deg,
                        float* __restrict__ ac, float* __restrict__ bb) {
    __shared__ float sx[16 * DPAD];
    const int tid  = threadIdx.x;
    const int row0 = blockIdx.x * 16;

    // stage 16 rows of node features into LDS, zero-padding K to DPAD
    for (int idx = tid; idx < 16 * DPAD; idx += 256) {
        const int r = idx / DPAD, k = idx - r * DPAD;
        sx[idx] = (k < DI) ? hin[(size_t)(row0 + r) * DI + k] : 0.0f;
    }
    __syncthreads();

    const int lane = tid & 31;
    const int wave = tid >> 5;          // 0..7 -> output column tile
    const int mrow = lane & 15;         // A row / B column within tile
    const int half = lane >> 4;         // upper half-wave holds K+2, K+3
    const int col  = wave * 16 + mrow;

    const v2f* __restrict__ p1 = reinterpret_cast<const v2f*>(P1);
    const v2f* __restrict__ p2 = reinterpret_cast<const v2f*>(P2);
    const v2f* __restrict__ p3 = reinterpret_cast<const v2f*>(P3);
    const float* __restrict__ sa = &sx[mrow * DPAD + half * 2];
    const int jb = half * 128 + col;

    v8f acc1 = {}, acc2 = {}, acc3 = {};
    #pragma unroll 4
    for (int k0 = 0; k0 < DPAD; k0 += 4) {
        const v2f A  = *reinterpret_cast<const v2f*>(sa + k0);   // ds_load_b64
        const int j  = (k0 >> 2) * 256 + jb;
        const v2f B1 = p1[j];                                    // global_load_b64
        const v2f B2 = p2[j];
        const v2f B3 = p3[j];
        acc1 = __builtin_amdgcn_wmma_f32_16x16x4_f32(false, A, false, B1, (short)0, acc1, false, false);
        acc2 = __builtin_amdgcn_wmma_f32_16x16x4_f32(false, A, false, B2, (short)0, acc2, false, false);
        acc3 = __builtin_amdgcn_wmma_f32_16x16x4_f32(false, A, false, B3, (short)0, acc3, false, false);
    }

    const float b1c  = b1[col];
    const float b3c  = b3[col];
    const int   rsel = half * 8;          // D: lanes 16-31 hold rows M+8
    #pragma unroll
    for (int jj = 0; jj < 8; ++jj) {
        const int row  = row0 + jj + rsel;
        const float av = acc1[jj] + b1c;
        const float cv = acc3[jj] + b3c;
        ac[(size_t)row * HID + col] = deg[row] * av + cv;
        bb[(size_t)row * HID + col] = acc2[jj];
    }
}

// --------------------------------------------- edge aggregation (the hot loop)
// One wave per edge: gather 128 floats of bb[src] (float4/lane), atomically
// subtract into ac[dst]. Both 51 MB arrays are L2-resident (192 MB L2).
__global__ __launch_bounds__(256)
void edge_agg_kernel(const long long* __restrict__ ei,
                     const float* __restrict__ bb, float* __restrict__ ac) {
    const int e    = (blockIdx.x * blockDim.x + threadIdx.x) >> 5;
    const int lane = threadIdx.x & 31;
    if (e >= N_EDGES) return;
    const int src = (int)ei[e];
    const int dst = (int)ei[N_EDGES + e];
    const float4 v = *reinterpret_cast<const float4*>(bb + (size_t)src * HID + lane * 4);
    float* d = ac + (size_t)dst * HID + lane * 4;
    atomicAdd(d + 0, -v.x);
    atomicAdd(d + 1, -v.y);
    atomicAdd(d + 2, -v.z);
    atomicAdd(d + 3, -v.w);
}

__global__ void relu_kernel(const float* __restrict__ in, float* __restrict__ out, int n) {
    int i = blockIdx.x * blockDim.x + threadIdx.x;
    if (i < n) out[i] = fmaxf(in[i], 0.0f);
}

// ------------------------------------------------------- global mean pooling
__global__ __launch_bounds__(256)
void pool_kernel(const float* __restrict__ h, const long long* __restrict__ batch,
                 float* __restrict__ pool, float* __restrict__ cnt) {
    const int node = (blockIdx.x * blockDim.x + threadIdx.x) >> 5;
    const int lane = threadIdx.x & 31;
    if (node >= N_NODES) return;
    const int g = (int)batch[node];
    const float4 v = *reinterpret_cast<const float4*>(h + (size_t)node * HID + lane * 4);
    float* p = pool + (size_t)g * HID + lane * 4;
    atomicAdd(p + 0, v.x);
    atomicAdd(p + 1, v.y);
    atomicAdd(p + 2, v.z);
    atomicAdd(p + 3, v.w);
    if (lane == 0) atomicAdd(&cnt[g], 1.0f);
}

__global__ void graph_mean_kernel(const float* __restrict__ pool,
                                  const float* __restrict__ cnt,
                                  float* __restrict__ gx) {
    int i = blockIdx.x * blockDim.x + threadIdx.x;
    if (i < N_GRAPHS * HID) {
        gx[i] = pool[i] / fmaxf(cnt[i / HID], 1.0f);
    }
}

// ----------------------------------------------------- FFN head, hidden layer
// relu(gx @ Wf1 + bf1), same fp32 WMMA tiling with packed B (2000 = 125 x 16).
__global__ __launch_bounds__(256)
void ffn1_kernel(const float* __restrict__ gx, const float* __restrict__ Pf1,
                 const float* __restrict__ bf1, float* __restrict__ hid) {
    __shared__ float sx[16 * HID];
    const int tid  = threadIdx.x;
    const int row0 = blockIdx.x * 16;
    for (int idx = tid; idx < 16 * HID; idx += 256)
        sx[idx] = gx[(size_t)row0 * HID + idx];
    __syncthreads();

    const int lane = tid & 31;
    const int wave = tid >> 5;
    const int mrow = lane & 15;
    const int half = lane >> 4;
    const int col  = wave * 16 + mrow;

    const v2f* __restrict__ p1 = reinterpret_cast<const v2f*>(Pf1);
    const float* __restrict__ sa = &sx[mrow * HID + half * 2];
    const int jb = half * 128 + col;

    v8f acc = {};
    #pragma unroll 4
    for (int k0 = 0; k0 < HID; k0 += 4) {
        const v2f A = *reinterpret_cast<const v2f*>(sa + k0);
        const v2f B = p1[(k0 >> 2) * 256 + jb];
        acc = __builtin_amdgcn_wmma_f32_16x16x4_f32(false, A, false, B, (short)0, acc, false, false);
    }
    const float bc   = bf1[col];
    const int   rsel = half * 8;
    #pragma unroll
    for (int jj = 0; jj < 8; ++jj) {
        const int row = row0 + jj + rsel;
        hid[(size_t)row * HID + col] = fmaxf(acc[jj] + bc, 0.0f);
    }
}

// Final tiny projection: [2000,128] @ [128,2] + bf2 — one thread per output.
__global__ void ffn2_kernel(const float* __restrict__ hid, const float* __restrict__ Wf2,
                            const float* __restrict__ bf2, float* __restrict__ out) {
    int i = blockIdx.x * blockDim.x + threadIdx.x;
    if (i >= N_GRAPHS * OUT_CLS) return;
    const int g = i / OUT_CLS, o = i - g * OUT_CLS;
    float s = bf2[o];
    #pragma unroll 4
    for (int c = 0; c < HID; ++c)
        s += hid[(size_t)g * HID + c] * Wf2[c * OUT_CLS + o];
    out[i] = s;
}

// ---------------------------------------------------------------------------
extern "C" void kernel_launch(void* const* d_in, const int* in_sizes, int n_in,
                              void* d_out, int out_size, void* d_ws, size_t ws_size,
                              hipStream_t stream) {
    (void)in_sizes; (void)n_in; (void)out_size; (void)ws_size;

    const float*      x     = (const float*)d_in[0];
    const long long*  ei    = (const long long*)d_in[1];
    const long long*  batch = (const long long*)d_in[2];
    const float* W1[3] = {(const float*)d_in[3],  (const float*)d_in[8],  (const float*)d_in[13]};
    const float* b1[3] = {(const float*)d_in[4],  (const float*)d_in[9],  (const float*)d_in[14]};
    const float* W2[3] = {(const float*)d_in[5],  (const float*)d_in[10], (const float*)d_in[15]};
    const float* W3[3] = {(const float*)d_in[6],  (const float*)d_in[11], (const float*)d_in[16]};
    const float* b3[3] = {(const float*)d_in[7],  (const float*)d_in[12], (const float*)d_in[17]};
    const float* Wf1 = (const float*)d_in[18];
    const float* bf1 = (const float*)d_in[19];
    const float* Wf2 = (const float*)d_in[20];
    const float* bf2 = (const float*)d_in[21];
    float* out = (float*)d_out;

    // workspace layout (all 16B aligned)
    const size_t NH  = (size_t)N_NODES * HID;
    const size_t PW0 = (size_t)DPAD0 * HID;   // packed layer-0 weight size
    const size_t PWH = (size_t)HID * HID;     // packed 128x128 weight size
    float* ws   = (float*)d_ws;
    float* h0   = ws;  ws += NH;
    float* h1   = ws;  ws += NH;
    float* ac   = ws;  ws += NH;                       // deg*a + c, then -= agg
    float* bb   = ws;  ws += NH;                       // message features x@W2
    float* deg  = ws;  ws += N_NODES;
    float* pool = ws;  ws += (size_t)N_GRAPHS * HID;   // cnt must follow pool
    float* cnt  = ws;  ws += N_GRAPHS;
    float* gx   = ws;  ws += (size_t)N_GRAPHS * HID;
    float* hid  = ws;  ws += (size_t)N_GRAPHS * HID;
    float* pw[3][3];
    for (int m = 0; m < 3; ++m) { pw[0][m] = ws; ws += PW0; }
    for (int l = 1; l < 3; ++l)
        for (int m = 0; m < 3; ++m) { pw[l][m] = ws; ws += PWH; }
    float* pwf1 = ws;  ws += PWH;

    // ---- pre-pack all weight matrices into WMMA B-fragment order
    {
        const int g0 = (DPAD0 * 64 + 255) / 256;
        const int gh = (HID * 64 + 255) / 256;
        pack_w_kernel<IN_DIM, DPAD0><<<g0, 256, 0, stream>>>(W1[0], pw[0][0]);
        pack_w_kernel<IN_DIM, DPAD0><<<g0, 256, 0, stream>>>(W2[0], pw[0][1]);
        pack_w_kernel<IN_DIM, DPAD0><<<g0, 256, 0, stream>>>(W3[0], pw[0][2]);
        for (int l = 1; l < 3; ++l) {
            pack_w_kernel<HID, HID><<<gh, 256, 0, stream>>>(W1[l], pw[l][0]);
            pack_w_kernel<HID, HID><<<gh, 256, 0, stream>>>(W2[l], pw[l][1]);
            pack_w_kernel<HID, HID><<<gh, 256, 0, stream>>>(W3[l], pw[l][2]);
        }
        pack_w_kernel<HID, HID><<<gh, 256, 0, stream>>>(Wf1, pwf1);
    }

    // ---- in-degree (edge weights are ones, identical for all 3 layers)
    zero_kernel<<<(N_NODES + 255) / 256, 256, 0, stream>>>(deg, N_NODES);
    deg_kernel<<<(N_EDGES + 255) / 256, 256, 0, stream>>>(ei, deg);

    // ---- layer 0 (di = 37)
    leconv_gemm_kernel<IN_DIM, DPAD0><<<N_NODES / 16, 256, 0, stream>>>(
        x, pw[0][0], b1[0], pw[0][1], pw[0][2], b3[0], deg, ac, bb);
    edge_agg_kernel<<<(N_EDGES * 32 + 255) / 256, 256, 0, stream>>>(ei, bb, ac);
    relu_kernel<<<(int)((NH + 255) / 256), 256, 0, stream>>>(ac, h0, (int)NH);

    // ---- layers 1, 2 (di = 128)
    const float* hin = h0;
    float* houts[2] = {h1, h0};
    for (int l = 1; l < 3; ++l) {
        leconv_gemm_kernel<HID, HID><<<N_NODES / 16, 256, 0, stream>>>(
            hin, pw[l][0], b1[l], pw[l][1], pw[l][2], b3[l], deg, ac, bb);
        edge_agg_kernel<<<(N_EDGES * 32 + 255) / 256, 256, 0, stream>>>(ei, bb, ac);
        relu_kernel<<<(int)((NH + 255) / 256), 256, 0, stream>>>(ac, houts[l - 1], (int)NH);
        hin = houts[l - 1];
    }

    // ---- global mean pool + FFN head
    zero_kernel<<<(N_GRAPHS * HID + N_GRAPHS + 255) / 256, 256, 0, stream>>>(
        pool, N_GRAPHS * HID + N_GRAPHS);
    pool_kernel<<<(N_NODES * 32 + 255) / 256, 256, 0, stream>>>(hin, batch, pool, cnt);
    graph_mean_kernel<<<(N_GRAPHS * HID + 255) / 256, 256, 0, stream>>>(pool, cnt, gx);
    ffn1_kernel<<<N_GRAPHS / 16, 256, 0, stream>>>(gx, pwf1, bf1, hid);
    ffn2_kernel<<<(N_GRAPHS * OUT_CLS + 255) / 256, 256, 0, stream>>>(hid, Wf2, bf2, out);
}